// SGConvNet_88811333746742
// MI455X (gfx1250) — compile-verified
//
#include <hip/hip_runtime.h>
#include <hip/hip_bf16.h>

#define NODES 100000
#define IN_C  128
#define HID   64
#define OUT_C 64

typedef __attribute__((ext_vector_type(2))) float v2f;
typedef __attribute__((ext_vector_type(8))) float v8f;

// ---------- degree / normalization ----------
__global__ void k_deg_init(float* __restrict__ deg, int n) {
    int i = blockIdx.x * blockDim.x + threadIdx.x;
    if (i < n) deg[i] = 1.0f;  // self-loop contributes 1 to every node's degree
}

__global__ void k_deg_accum(const long long* __restrict__ dst, int E,
                            float* __restrict__ deg) {
    int e = blockIdx.x * blockDim.x + threadIdx.x;
    if (e < E) atomicAdd(&deg[(int)dst[e]], 1.0f);
}

__global__ void k_dinv(float* __restrict__ deg, int n) {
    int i = blockIdx.x * blockDim.x + threadIdx.x;
    if (i < n) deg[i] = rsqrtf(deg[i]);  // deg >= 1 always (self loop)
}

// ---------- GEMM1: H = relu(X @ W1 + b1), f32 WMMA 16x16x4 ----------
// block = 128 threads (4 waves); wave w owns column tile w; block owns 16 rows.
__global__ void k_gemm1_relu(const float* __restrict__ X,
                             const float* __restrict__ W,   // [128,64] row-major
                             const float* __restrict__ bias,
                             float* __restrict__ H) {
    const int lane    = threadIdx.x & 31;
    const int wave    = threadIdx.x >> 5;          // 0..3
    const int rowBase = blockIdx.x * 16;
    const int colBase = wave * 16;
    const int m       = lane & 15;                  // M index / N index
    const int kh      = lane >> 4;                  // K half (0 or 1)

    const float* arow = X + (long)(rowBase + m) * IN_C + kh * 2;
    const float* bcol = W + colBase + m;

    v8f c = {};
#pragma unroll
    for (int k0 = 0; k0 < IN_C; k0 += 4) {
        // A 16x4 f32 fragment: M=lane%16, K = (lane/16)*2 + v  -> contiguous pair
        v2f a = *(const v2f*)(arow + k0);
        // B 4x16 f32 fragment: N=lane%16, K = (lane/16)*2 + v
        const int kr = k0 + kh * 2;
        v2f b;
        b.x = bcol[(long)kr * HID];
        b.y = bcol[(long)(kr + 1) * HID];
        c = __builtin_amdgcn_wmma_f32_16x16x4_f32(false, a, false, b,
                                                  (short)0, c, false, false);
    }
    const float bb = bias[colBase + m];
#pragma unroll
    for (int v = 0; v < 8; v++) {
        const int r = rowBase + v + kh * 8;          // C/D layout: M = v + (lane/16)*8
        float val = fmaxf(c[v] + bb, 0.0f);
        H[(long)r * HID + colBase + m] = val;
    }
}

// ---------- GEMM2: OUT = H @ W2 + b2, f32 WMMA 16x16x4 ----------
__global__ void k_gemm2(const float* __restrict__ Hin,
                        const float* __restrict__ W,   // [64,64]
                        const float* __restrict__ bias,
                        float* __restrict__ Out) {
    const int lane    = threadIdx.x & 31;
    const int wave    = threadIdx.x >> 5;
    const int rowBase = blockIdx.x * 16;
    const int colBase = wave * 16;
    const int m       = lane & 15;
    const int kh      = lane >> 4;

    const float* arow = Hin + (long)(rowBase + m) * HID + kh * 2;
    const float* bcol = W + colBase + m;

    v8f c = {};
#pragma unroll
    for (int k0 = 0; k0 < HID; k0 += 4) {
        v2f a = *(const v2f*)(arow + k0);
        const int kr = k0 + kh * 2;
        v2f b;
        b.x = bcol[(long)kr * OUT_C];
        b.y = bcol[(long)(kr + 1) * OUT_C];
        c = __builtin_amdgcn_wmma_f32_16x16x4_f32(false, a, false, b,
                                                  (short)0, c, false, false);
    }
    const float bb = bias[colBase + m];
#pragma unroll
    for (int v = 0; v < 8; v++) {
        const int r = rowBase + v + kh * 8;
        Out[(long)r * OUT_C + colBase + m] = c[v] + bb;
    }
}

// ---------- propagation ----------
// hout[i] = dinv[i]^2 * hin[i]   (self-loop term, also zero-initializes target)
__global__ void k_self_scale(const float* __restrict__ hin,
                             const float* __restrict__ dinv,
                             float* __restrict__ hout, int total) {
    int i = blockIdx.x * blockDim.x + threadIdx.x;
    if (i < total) {
        const float s = dinv[i >> 6];  // node = i / 64
        hout[i] = s * s * hin[i];
    }
}

// one wave32 per edge; each lane handles 2 of the 64 channels
__global__ void k_edge_scatter(const long long* __restrict__ src,
                               const long long* __restrict__ dst,
                               const float* __restrict__ dinv,
                               const float* __restrict__ hin,
                               float* __restrict__ hout, int E) {
    const unsigned gid  = blockIdx.x * blockDim.x + threadIdx.x;
    const unsigned e    = gid >> 5;
    const unsigned lane = gid & 31;
    if (e >= E) return;  // uniform per wave (wave32)
    const int s = (int)src[e];
    const int d = (int)dst[e];
    const float nrm = dinv[s] * dinv[d];
    const v2f hv = *(const v2f*)(hin + (long)s * HID + lane * 2);
    float* out = hout + (long)d * HID + lane * 2;
    atomicAdd(out,     nrm * hv.x);
    atomicAdd(out + 1, nrm * hv.y);
}

extern "C" void kernel_launch(void* const* d_in, const int* in_sizes, int n_in,
                              void* d_out, int out_size, void* d_ws, size_t ws_size,
                              hipStream_t stream) {
    const float*     x  = (const float*)d_in[0];
    const long long* ei = (const long long*)d_in[1];   // int64 [2, E]
    const float*     W1 = (const float*)d_in[2];
    const float*     b1 = (const float*)d_in[3];
    const float*     W2 = (const float*)d_in[4];
    const float*     b2 = (const float*)d_in[5];

    const int E = in_sizes[1] / 2;
    const long long* src = ei;
    const long long* dst = ei + E;

    float* h0   = (float*)d_ws;                    // NODES*64 floats
    float* dinv = h0 + (size_t)NODES * HID;        // NODES floats
    float* h1   = (float*)d_out;                   // reuse output as ping-pong buffer

    const int total = NODES * HID;

    // normalization
    k_deg_init <<<(NODES + 255) / 256, 256, 0, stream>>>(dinv, NODES);
    k_deg_accum<<<(E + 255) / 256,     256, 0, stream>>>(dst, E, dinv);
    k_dinv     <<<(NODES + 255) / 256, 256, 0, stream>>>(dinv, NODES);

    // h0 = relu(x @ W1 + b1)
    k_gemm1_relu<<<NODES / 16, 128, 0, stream>>>(x, W1, b1, h0);

    // hop 1: h1 = A_norm @ h0
    k_self_scale<<<(total + 255) / 256, 256, 0, stream>>>(h0, dinv, h1, total);
    k_edge_scatter<<<(unsigned)(((size_t)E * 32 + 255) / 256), 256, 0, stream>>>(
        src, dst, dinv, h0, h1, E);

    // hop 2: h0 = A_norm @ h1
    k_self_scale<<<(total + 255) / 256, 256, 0, stream>>>(h1, dinv, h0, total);
    k_edge_scatter<<<(unsigned)(((size_t)E * 32 + 255) / 256), 256, 0, stream>>>(
        src, dst, dinv, h1, h0, E);

    // out = h0 @ W2 + b2  (fully overwrites d_out)
    k_gemm2<<<NODES / 16, 128, 0, stream>>>(h0, W2, b2, (float*)d_out);
}